// CRF_64218351009815
// MI455X (gfx1250) — compile-verified
//
#include <hip/hip_runtime.h>
#include <cstdint>
#include <cstddef>

// Problem sizes (fixed by the reference)
#define Bn 64
#define Tn 512
#define Dn 1024
#define Un 128

typedef float v2f __attribute__((ext_vector_type(2)));
typedef float v8f __attribute__((ext_vector_type(8)));

// ---------------------------------------------------------------------------
// Kernel 1: logits = inputs @ kernel + bias  via V_WMMA_F32_16X16X4_F32
//   grid: 256 blocks x 256 threads (8 waves). Each wave -> one 16x128 tile.
//   Weights staged per-64-row K-chunk in LDS, pre-swizzled as [k4][col][kr]
//   so each lane's B fragment is a single contiguous ds_load_b64.
// ---------------------------------------------------------------------------
__global__ __launch_bounds__(256) void crf_logits_gemm(
    const float* __restrict__ in,    // [B*T, D]
    const float* __restrict__ wk,    // [D, U]
    const float* __restrict__ bias,  // [U]
    float* __restrict__ logits)      // [B*T, U]
{
  // [k4 = 0..15][col = 0..127][kr = 0..3]  (32 KB)
  __shared__ float lds_b[16 * 128 * 4];

  const int tid  = threadIdx.x;
  const int lane = tid & 31;
  const int wid  = tid >> 5;
  const int mt   = blockIdx.x * 8 + wid;   // 0..2047 M-tiles
  const int m0   = mt * 16;
  const int mrow = m0 + (lane & 15);
  const int klo  = (lane >> 4) << 1;       // 0 for lanes 0-15, 2 for lanes 16-31

  v8f acc[8] = {};                          // 8 N-tiles of 16x16 f32

  for (int k0 = 0; k0 < Dn; k0 += 64) {
    __syncthreads();
    // Stage wk[k0..k0+64) x [0..128) into swizzled LDS.
    // Thread handles (k4,col): 4 coalesced global b32 loads + 1 float4 LDS store.
#pragma unroll
    for (int p = 0; p < 8; ++p) {
      const int q   = p * 256 + tid;        // 0..2047
      const int k4  = q >> 7;               // 0..15
      const int col = q & 127;
      const int row = k0 + (k4 << 2);
      float4 v;
      v.x = wk[(size_t)(row + 0) * Un + col];
      v.y = wk[(size_t)(row + 1) * Un + col];
      v.z = wk[(size_t)(row + 2) * Un + col];
      v.w = wk[(size_t)(row + 3) * Un + col];
      *(float4*)(&lds_b[((k4 << 7) + col) << 2]) = v;
    }
    __syncthreads();
    if (k0 + 64 < Dn)  // pull next weight chunk toward L2 (global_prefetch_b8)
      __builtin_prefetch(wk + (size_t)(k0 + 64) * Un + (size_t)tid * 16, 0, 0);

#pragma unroll 4
    for (int kk = 0; kk < 64; kk += 4) {
      // A fragment: 16x4 fp32, one b64 per lane.
      // Lane row = m0+(lane&15); VGPR0/1 = K{klo,klo+1}.
      const float2 a2 = *(const float2*)(in + (size_t)mrow * Dn + (k0 + kk + klo));
      v2f a; a.x = a2.x; a.y = a2.y;
      const int kbase = ((kk >> 2) << 9);   // (kk/4)*128*4
#pragma unroll
      for (int nt = 0; nt < 8; ++nt) {
        const int col = nt * 16 + (lane & 15);
        // Contiguous pair {K=klo, K=klo+1} -> single ds_load_b64, bank-conflict-free.
        const float2 b2 = *(const float2*)(&lds_b[kbase + (col << 2) + klo]);
        v2f bf; bf.x = b2.x; bf.y = b2.y;
        // (neg_a, A, neg_b, B, c_mod, C, reuse_a, reuse_b)
        acc[nt] = __builtin_amdgcn_wmma_f32_16x16x4_f32(
            false, a, false, bf, (short)0, acc[nt], false, false);
      }
    }
  }

  // Epilogue: C/D layout -> VGPR v holds row m0+v (lanes 0-15) / m0+8+v (16-31)
  const int rbase = m0 + ((lane >> 4) << 3);
#pragma unroll
  for (int nt = 0; nt < 8; ++nt) {
    const int col = nt * 16 + (lane & 15);
    const float bv = bias[col];
#pragma unroll
    for (int v = 0; v < 8; ++v)
      logits[(size_t)(rbase + v) * Un + col] = acc[nt][v] + bv;
  }
}

// ---------------------------------------------------------------------------
// Kernel 2: Viterbi forward pass. One block per batch element, thread = tag j.
//   new_state[j] = pot_t[j] + max_i(state[i] + chain[i][j]); backptr = argmax_i
// ---------------------------------------------------------------------------
__global__ __launch_bounds__(128) void crf_viterbi_fwd(
    const float* __restrict__ logits,   // [B, T, U]
    const float* __restrict__ chain,    // [U, U]
    uint8_t* __restrict__ bp,           // [B, T-1, U]
    float* __restrict__ finalst)        // [B, U]
{
  const int b = blockIdx.x;
  const int j = threadIdx.x;
  __shared__ float st[2][Un];

  st[0][j] = logits[(size_t)b * Tn * Un + j];
  __syncthreads();

  for (int t = 1; t < Tn; ++t) {
    const float* cur = st[(t - 1) & 1];
    float best = -3.402823466e38f;
    int   arg  = 0;
#pragma unroll 8
    for (int i = 0; i < Un; ++i) {
      const float s = cur[i] + chain[i * Un + j];  // coalesced in j
      if (s > best) { best = s; arg = i; }         // strict '>' == first-max (jnp.argmax)
    }
    bp[((size_t)b * (Tn - 1) + (t - 1)) * Un + j] = (uint8_t)arg;
    st[t & 1][j] = logits[((size_t)b * Tn + t) * Un + j] + best;
    __syncthreads();  // iter t's reads (buf (t-1)&1) done before iter t+1 overwrites it
  }
  finalst[b * Un + j] = st[(Tn - 1) & 1][j];
}

// ---------------------------------------------------------------------------
// Kernel 3: backtrace. One thread per batch element.
// ---------------------------------------------------------------------------
__global__ void crf_backtrace(const float* __restrict__ finalst,
                              const uint8_t* __restrict__ bp,
                              float* __restrict__ out)   // [B, T] as f32
{
  const int b = blockIdx.x * blockDim.x + threadIdx.x;
  if (b >= Bn) return;
  int tag = 0;
  float best = finalst[b * Un];
  for (int j = 1; j < Un; ++j) {
    const float v = finalst[b * Un + j];
    if (v > best) { best = v; tag = j; }   // first-max tie-break
  }
  out[(size_t)b * Tn + (Tn - 1)] = (float)tag;
  for (int s = Tn - 2; s >= 0; --s) {
    tag = bp[((size_t)b * (Tn - 1) + s) * Un + tag];
    out[(size_t)b * Tn + s] = (float)tag;
  }
}

// ---------------------------------------------------------------------------
extern "C" void kernel_launch(void* const* d_in, const int* in_sizes, int n_in,
                              void* d_out, int out_size, void* d_ws, size_t ws_size,
                              hipStream_t stream) {
  (void)in_sizes; (void)n_in; (void)out_size; (void)ws_size;
  const float* inputs = (const float*)d_in[0];   // (B,T,D) f32
  const float* wk     = (const float*)d_in[1];   // (D,U)   f32
  const float* bias   = (const float*)d_in[2];   // (U,)    f32
  const float* chain  = (const float*)d_in[3];   // (U,U)   f32
  float* out = (float*)d_out;                    // (B,T)   f32

  // Workspace layout: logits (16 MB) | backptrs u8 (4.19 MB) | final state (32 KB)
  const size_t LOGITS_BYTES = (size_t)Bn * Tn * Un * sizeof(float);
  const size_t BP_BYTES     = (size_t)Bn * (Tn - 1) * Un;
  float*   logits  = (float*)d_ws;
  uint8_t* bp      = (uint8_t*)d_ws + LOGITS_BYTES;
  float*   finalst = (float*)((char*)d_ws + LOGITS_BYTES + BP_BYTES);

  crf_logits_gemm<<<dim3(256), dim3(256), 0, stream>>>(inputs, wk, bias, logits);
  crf_viterbi_fwd<<<dim3(Bn), dim3(Un), 0, stream>>>(logits, chain, bp, finalst);
  crf_backtrace<<<dim3(1), dim3(Bn), 0, stream>>>(finalst, bp, out);
}